// FourierCrossAttentionN_27169963114629
// MI455X (gfx1250) — compile-verified
//
#include <hip/hip_runtime.h>
#include <math.h>

// FourierCrossAttention for MI455X (gfx1250, wave32, WMMA).
//
// Shapes: B=8,H=8,E=O=16, S=(32,32,64), modes 8x8x8=512.
// Pipeline (all fp32 / complex64, matching reference precision):
//   P1: per (b,h,e,x) plane (32y x 64z) -> partial DFT -> T2[bhe][x][64 modes]
//   P2: contract x (32->8)              -> Xsel[bhe][512]         (x ortho 1/256)
//   ATT: fused tanh(Q^T K) * K^T via V_WMMA_F32_16X16X4_F32; att never hits memory
//   W : per-mode 16x16 complex matvec with (w_re + i w_im), /16384
//   I1: Re{ ifft2 } over (m0,m1) -> P[bho][x][y][8] real          (x 1/32)
//   I2: irfft_z expansion (cosine series) -> out, streams 256MB once (x 1/8)
//
// Bandwidth plan: q/k (512MB) and out (256MB) are touch-once streams > 192MB L2,
// so they use non-temporal loads/stores; the small reused intermediates
// (Xq/Xk/xqkv/xqkvw, 16MB) stay regular-temporal and live in L2.
//
// Workspace layout (64 MB):
//   [0,16M)  T2 (reused for q then k), [16M,20M) Xq, [20M,24M) Xk,
//   [24M,28M) xqkv, [28M,32M) xqkvw, [32M,64M) P

typedef float v2f __attribute__((ext_vector_type(2)));
typedef float v4f __attribute__((ext_vector_type(4)));
typedef float v8f __attribute__((ext_vector_type(8)));

#define PI2 6.28318530717958647692f

__device__ inline v8f wmma_f32(v2f a, v2f b, v8f c) {
    // D(16x16,f32) = A(16x4,f32) * B(4x16,f32) + C
    return __builtin_amdgcn_wmma_f32_16x16x4_f32(false, a, false, b, (short)0, c,
                                                 false, false);
}

// ---------------------------------------------------------------- forward y,z
__global__ __launch_bounds__(256) void k_fwd_plane(const float* __restrict__ src,
                                                   float2* __restrict__ T2)
{
    __shared__ float  plane[32 * 65];   // padded rows -> no bank conflicts
    __shared__ float2 tw2s[8 * 65];     // e^{-2pi i m2 z / 64}, padded
    __shared__ float2 tw1s[256];        // e^{-2pi i m1 y / 32}
    __shared__ float2 Sz[256];          // [y][m2]
    const int t = threadIdx.x;
    const size_t planeIdx = blockIdx.x;              // bhe*32 + x
    const v4f* p4 = (const v4f*)(src + planeIdx * 2048);
#pragma unroll
    for (int i = 0; i < 2; ++i) {                    // load 2048 floats, NT (read-once)
        int idx4 = t + i * 256;
        v4f v = __builtin_nontemporal_load(&p4[idx4]);
        int base = idx4 * 4;
        int y = base >> 6, z = base & 63;
        float* dst = &plane[y * 65 + z];
        dst[0] = v[0]; dst[1] = v[1]; dst[2] = v[2]; dst[3] = v[3];
    }
#pragma unroll
    for (int i = 0; i < 2; ++i) {
        int idx = t + i * 256;
        int m2 = idx >> 6, z = idx & 63;
        float s, c; __sincosf(PI2 * (float)(m2 * z) * (1.0f / 64.0f), &s, &c);
        tw2s[m2 * 65 + z] = make_float2(c, -s);
    }
    {
        int m1 = t >> 5, y = t & 31;
        float s, c; __sincosf(PI2 * (float)(m1 * y) * (1.0f / 32.0f), &s, &c);
        tw1s[t] = make_float2(c, -s);
    }
    __syncthreads();
    {   // z-contraction: thread = (y, m2)
        int y = t >> 3, m2 = t & 7;
        float re = 0.f, im = 0.f;
#pragma unroll
        for (int z = 0; z < 64; ++z) {
            float v = plane[y * 65 + z];
            float2 w = tw2s[m2 * 65 + z];
            re = fmaf(v, w.x, re);
            im = fmaf(v, w.y, im);
        }
        Sz[y * 8 + m2] = make_float2(re, im);
    }
    __syncthreads();
    if (t < 64) {   // y-contraction: thread = (m1, m2); mode = m1*8+m2 = t
        int m1 = t >> 3, m2 = t & 7;
        float re = 0.f, im = 0.f;
#pragma unroll
        for (int y = 0; y < 32; ++y) {
            float2 s = Sz[y * 8 + m2];
            float2 w = tw1s[m1 * 32 + y];
            re += s.x * w.x - s.y * w.y;
            im += s.x * w.y + s.y * w.x;
        }
        T2[planeIdx * 64 + t] = make_float2(re, im);
    }
}

// ------------------------------------------------------------------ forward x
__global__ __launch_bounds__(256) void k_fwd_x(const float2* __restrict__ T2,
                                               float2* __restrict__ X)
{
    __shared__ float2 tw0[256];
    const int t = threadIdx.x;
    {
        int m0 = t >> 5, x = t & 31;
        float s, c; __sincosf(PI2 * (float)(m0 * x) * (1.0f / 32.0f), &s, &c);
        tw0[t] = make_float2(c, -s);
    }
    __syncthreads();
    size_t g = (size_t)blockIdx.x * 256 + t;         // 0 .. 524287
    int bhe = (int)(g >> 9);
    int r = (int)(g & 511);
    int m0 = r >> 6, mm = r & 63;
    const float2* row = T2 + (size_t)bhe * 2048;
    float re = 0.f, im = 0.f;
#pragma unroll
    for (int x = 0; x < 32; ++x) {
        float2 v = row[x * 64 + mm];
        float2 w = tw0[m0 * 32 + x];
        re += v.x * w.x - v.y * w.y;
        im += v.x * w.y + v.y * w.x;
    }
    X[g] = make_float2(re * (1.0f / 256.0f), im * (1.0f / 256.0f)); // rfftn ortho
}

// ----------------------------------------------------- fused complex attention
// block = 256 thr = 8 waves; wave owns one 16-wide x-tile of one (b,h).
// grid = 64 bh * 4 quarters = 256 blocks.
__global__ __launch_bounds__(256) void k_attention(const float2* __restrict__ Xq,
                                                   const float2* __restrict__ Xk,
                                                   float2* __restrict__ xqkv)
{
    __shared__ __align__(16) float2 smem[8][512];    // per wave: Ks[256] | att[256]
    const int t = threadIdx.x;
    const int wave = t >> 5;
    const int lane = t & 31;
    const int half = (lane >> 4) & 1;
    const int l16 = lane & 15;

    const int bh = blockIdx.x >> 2;
    const int xtile = (blockIdx.x & 3) * 8 + wave;

    float2* Ks = smem[wave];
    float2* As = smem[wave] + 256;

    // Q tile in A layout: element (M=x=l16, K=e = ec*4 + half*2 + v)
    float2 qa[4][2];
#pragma unroll
    for (int ec = 0; ec < 4; ++ec)
#pragma unroll
        for (int v = 0; v < 2; ++v) {
            int e = ec * 4 + half * 2 + v;
            qa[ec][v] = Xq[((size_t)bh * 16 + e) * 512 + xtile * 16 + l16];
        }

    v8f acc_rr = {}, acc_ri = {}, acc_ir = {}, acc_ii = {};

    for (int yt = 0; yt < 32; ++yt) {
        {   // stage K tile (16e x 16y) -> LDS, coalesced b128
            int e = lane >> 1;
            int y0 = (lane & 1) * 8;
            const float4* g = (const float4*)(Xk + ((size_t)bh * 16 + e) * 512 +
                                              yt * 16 + y0);
            float4* d = (float4*)(Ks + e * 16 + y0);
#pragma unroll
            for (int j = 0; j < 4; ++j) d[j] = g[j];
        }
        asm volatile("" ::: "memory");   // wave-private LDS, DS pipe is in-order

        // GEMM1: att[x,y] = sum_e Q[e,x] K[e,y]  (complex via 4 real WMMAs)
        v8f c_rr = {}, c_ri = {}, c_ir = {}, c_ii = {};
#pragma unroll
        for (int ec = 0; ec < 4; ++ec) {
            v2f a_re, a_im, b_re, b_im;
            a_re[0] = qa[ec][0].x; a_re[1] = qa[ec][1].x;
            a_im[0] = qa[ec][0].y; a_im[1] = qa[ec][1].y;
#pragma unroll
            for (int v = 0; v < 2; ++v) {
                int e = ec * 4 + half * 2 + v;
                float2 kv = Ks[e * 16 + l16];
                b_re[v] = kv.x; b_im[v] = kv.y;
            }
            c_rr = wmma_f32(a_re, b_re, c_rr);
            c_ii = wmma_f32(a_im, b_im, c_ii);
            c_ri = wmma_f32(a_re, b_im, c_ri);
            c_ir = wmma_f32(a_im, b_re, c_ir);
        }
        // complex tanh, write att tile to LDS in (x,y) order for GEMM2's A
#pragma unroll
        for (int vp = 0; vp < 8; ++vp) {
            float ar = c_rr[vp] - c_ii[vp];
            float ai = c_ri[vp] + c_ir[vp];
            float a2 = fminf(fmaxf(2.0f * ar, -30.0f), 30.0f);
            float ea = __expf(a2);
            float ie = 1.0f / ea;
            float sh = 0.5f * (ea - ie);
            float ch = 0.5f * (ea + ie);
            float sb, cb; __sincosf(2.0f * ai, &sb, &cb);
            float inv = 1.0f / (ch + cb);
            int xin = vp + 8 * half;                 // D layout: M = vgpr + 8*half
            As[xin * 16 + l16] = make_float2(sh * inv, sb * inv);
        }
        asm volatile("" ::: "memory");

        // GEMM2: xqkv[x,e] += sum_y att[x,y] K[e,y]
#pragma unroll
        for (int kc = 0; kc < 4; ++kc) {
            v2f a_re, a_im, b_re, b_im;
#pragma unroll
            for (int v = 0; v < 2; ++v) {
                int yk = kc * 4 + half * 2 + v;
                float2 av = As[l16 * 16 + yk];       // A: (M=x=l16, K=y)
                float2 bv = Ks[l16 * 16 + yk];       // B: (N=e=l16, K=y)
                a_re[v] = av.x; a_im[v] = av.y;
                b_re[v] = bv.x; b_im[v] = bv.y;
            }
            acc_rr = wmma_f32(a_re, b_re, acc_rr);
            acc_ii = wmma_f32(a_im, b_im, acc_ii);
            acc_ri = wmma_f32(a_re, b_im, acc_ri);
            acc_ir = wmma_f32(a_im, b_re, acc_ir);
        }
    }
#pragma unroll
    for (int vp = 0; vp < 8; ++vp) {                 // D: (M -> x, N=l16 -> e)
        float re = acc_rr[vp] - acc_ii[vp];
        float im = acc_ri[vp] + acc_ir[vp];
        int x = xtile * 16 + vp + 8 * half;
        xqkv[((size_t)bh * 16 + l16) * 512 + x] = make_float2(re, im);
    }
}

// ---------------------------------------------------------- per-mode weights
__global__ __launch_bounds__(256) void k_weights(const float2* __restrict__ xqkv,
                                                 const float* __restrict__ w_re,
                                                 const float* __restrict__ w_im,
                                                 float2* __restrict__ xqkvw)
{
    size_t g = (size_t)blockIdx.x * 256 + threadIdx.x;   // 0..524287
    int bh = (int)(g >> 13);
    int rem = (int)(g & 8191);
    int o = rem >> 9;
    int mode = rem & 511;
    int h = bh & 7;
    float re = 0.f, im = 0.f;
#pragma unroll
    for (int e = 0; e < 16; ++e) {
        float2 xv = xqkv[((size_t)bh * 16 + e) * 512 + mode];
        size_t wi = (((size_t)h * 16 + e) * 16 + o) * 512 + mode;
        float wr = w_re[wi], wim = w_im[wi];
        re += xv.x * wr - xv.y * wim;
        im += xv.x * wim + xv.y * wr;
    }
    const float s = 1.0f / 16384.0f;                 // /(C_IN*C_OUT)
    xqkvw[((size_t)bh * 16 + o) * 512 + mode] = make_float2(re * s, im * s);
}

// --------------------------- inverse over (m0,m1): canonicalize == Re{ifft2}
__global__ __launch_bounds__(256) void k_inv_xy(const float2* __restrict__ xqkvw,
                                                float* __restrict__ P)
{
    __shared__ float2 Af[512];
    __shared__ float2 twi[256];                      // e^{+2pi i m x / 32}
    const int t = threadIdx.x;
    const size_t bho = blockIdx.x;
    Af[t] = xqkvw[bho * 512 + t];
    Af[t + 256] = xqkvw[bho * 512 + t + 256];
    {
        int m = t >> 5, x = t & 31;
        float s, c; __sincosf(PI2 * (float)(m * x) * (1.0f / 32.0f), &s, &c);
        twi[t] = make_float2(c, s);
    }
    __syncthreads();
    int y = t >> 3, m2 = t & 7;
    float2 U[8];
#pragma unroll
    for (int m0 = 0; m0 < 8; ++m0) {
        float re = 0.f, im = 0.f;
#pragma unroll
        for (int m1 = 0; m1 < 8; ++m1) {
            float2 a = Af[m0 * 64 + m1 * 8 + m2];
            float2 w = twi[m1 * 32 + y];
            re += a.x * w.x - a.y * w.y;
            im += a.x * w.y + a.y * w.x;
        }
        U[m0] = make_float2(re, im);
    }
    float* dst = P + bho * 8192;
    for (int x = 0; x < 32; ++x) {
        float s = 0.f;
#pragma unroll
        for (int m0 = 0; m0 < 8; ++m0) {
            float2 w = twi[m0 * 32 + x];
            s += U[m0].x * w.x - U[m0].y * w.y;      // real part only
        }
        dst[x * 256 + y * 8 + m2] = s * 0.03125f;    // ifft2 ortho 1/32
    }
}

// ------------------------------- irfft_z cosine expansion, streams out once
__global__ __launch_bounds__(256) void k_inv_z(const float* __restrict__ P,
                                               float* __restrict__ out)
{
    __shared__ float w2s[512];
    const int t = threadIdx.x;
#pragma unroll
    for (int i = 0; i < 2; ++i) {
        int idx = t + i * 256;
        int m2 = idx >> 6, z = idx & 63;
        float c = cosf(PI2 * (float)(m2 * z) * (1.0f / 64.0f));
        w2s[idx] = 0.125f * (m2 ? 2.0f : 1.0f) * c;  // irfft ortho 1/sqrt(64)
    }
    __syncthreads();
    size_t row = (size_t)blockIdx.x * 256 + t;       // (bho, x, y) row
    const v4f* pr = (const v4f*)(P + row * 8);
    v4f p0 = __builtin_nontemporal_load(&pr[0]);     // P is read-once
    v4f p1 = __builtin_nontemporal_load(&pr[1]);
    float pv[8] = {p0[0], p0[1], p0[2], p0[3], p1[0], p1[1], p1[2], p1[3]};
    v4f* o = (v4f*)(out + row * 64);
#pragma unroll
    for (int zq = 0; zq < 16; ++zq) {
        v4f r;
#pragma unroll
        for (int j = 0; j < 4; ++j) {
            int z = zq * 4 + j;
            float s = 0.f;
#pragma unroll
            for (int m2 = 0; m2 < 8; ++m2) s = fmaf(pv[m2], w2s[m2 * 64 + z], s);
            r[j] = s;
        }
        __builtin_nontemporal_store(r, &o[zq]);      // out is write-once, 256MB
    }
}

// ------------------------------------------------------------------- launcher
extern "C" void kernel_launch(void* const* d_in, const int* in_sizes, int n_in,
                              void* d_out, int out_size, void* d_ws, size_t ws_size,
                              hipStream_t stream)
{
    (void)in_sizes; (void)n_in; (void)out_size; (void)ws_size;
    const float* q = (const float*)d_in[0];
    const float* k = (const float*)d_in[1];
    const float* w_re = (const float*)d_in[2];
    const float* w_im = (const float*)d_in[3];
    float* out = (float*)d_out;
    char* ws = (char*)d_ws;

    float2* T2    = (float2*)(ws);                          // 16 MB (reused)
    float2* Xq    = (float2*)(ws + (size_t)(16u << 20));    //  4 MB
    float2* Xk    = (float2*)(ws + (size_t)(20u << 20));    //  4 MB
    float2* xqkv  = (float2*)(ws + (size_t)(24u << 20));    //  4 MB
    float2* xqkvw = (float2*)(ws + (size_t)(28u << 20));    //  4 MB
    float*  P     = (float*)(ws + (size_t)(32u << 20));     // 32 MB

    hipLaunchKernelGGL(k_fwd_plane, dim3(32768), dim3(256), 0, stream, q, T2);
    hipLaunchKernelGGL(k_fwd_x,     dim3(2048),  dim3(256), 0, stream, T2, Xq);
    hipLaunchKernelGGL(k_fwd_plane, dim3(32768), dim3(256), 0, stream, k, T2);
    hipLaunchKernelGGL(k_fwd_x,     dim3(2048),  dim3(256), 0, stream, T2, Xk);
    hipLaunchKernelGGL(k_attention, dim3(256),   dim3(256), 0, stream, Xq, Xk, xqkv);
    hipLaunchKernelGGL(k_weights,   dim3(2048),  dim3(256), 0, stream, xqkv, w_re, w_im, xqkvw);
    hipLaunchKernelGGL(k_inv_xy,    dim3(1024),  dim3(256), 0, stream, xqkvw, P);
    hipLaunchKernelGGL(k_inv_z,     dim3(4096),  dim3(256), 0, stream, P, out);
}